// GTModel_40080634806432
// MI455X (gfx1250) — compile-verified
//
#include <hip/hip_runtime.h>
#include <math.h>

#define N_NODES 20000
#define N_EDGES 320000
#define DIM 64
#define NH 4
#define HDIM 256
#define NG 64
#define NL 3
#define DENSE 256
#define BN_EPS 1e-5f

typedef __attribute__((ext_vector_type(2))) float v2f;
typedef __attribute__((ext_vector_type(8))) float v8f;
// pointee type expected by the async-LDS builtin (matches clang's diagnostic)
typedef int v4i_ld __attribute__((vector_size(4 * sizeof(int))));

#ifndef __has_builtin
#define __has_builtin(x) 0
#endif

#if __has_builtin(__builtin_amdgcn_global_load_async_to_lds_b128)
#define HAVE_ASYNC_LDS 1
#else
#define HAVE_ASYNC_LDS 0
#endif

// ---- order-preserving float<->uint keys for atomicMax on signed floats ----
__device__ __forceinline__ unsigned fmax_key(float f) {
    unsigned u = __float_as_uint(f);
    return u ^ (unsigned)(((int)u >> 31) | 0x80000000);
}
__device__ __forceinline__ float fmax_unkey(unsigned u) {
    unsigned v = (u & 0x80000000u) ? (u ^ 0x80000000u) : ~u;
    return __uint_as_float(v);
}
#define NEG_INF_KEY 0x007FFFFFu   // fmax_key(-inf)

// =====================================================================
// Generic fill
// =====================================================================
__global__ void fill_u32(unsigned* __restrict__ p, unsigned v, int n) {
    int i = blockIdx.x * blockDim.x + threadIdx.x;
    if (i < n) p[i] = v;
}

__global__ void count_kernel(const int* __restrict__ batch, int* __restrict__ counts) {
    int i = blockIdx.x * blockDim.x + threadIdx.x;
    if (i < N_NODES) atomicAdd(&counts[batch[i]], 1);
}

// =====================================================================
// Node GEMM: O = x @ W + b  for 4 weight sets (Q,K,V,Skip)
// x:[N,64]  W:[64,256]  O:[N,256]
// block=128 (4 waves), grid=(N/16, 4, 4); wave owns a 16x16 C tile.
// =====================================================================
__global__ void qkvs_gemm(const float* __restrict__ x,
                          const float* __restrict__ Wq, const float* __restrict__ bq,
                          const float* __restrict__ Wk, const float* __restrict__ bk,
                          const float* __restrict__ Wv, const float* __restrict__ bv,
                          const float* __restrict__ Ws, const float* __restrict__ bs,
                          float* __restrict__ Q, float* __restrict__ K,
                          float* __restrict__ V, float* __restrict__ S) {
    const int lane = threadIdx.x & 31;
    const int wave = threadIdx.x >> 5;
    const int n = lane & 15;
    const int hi = lane >> 4;
    const int r0 = blockIdx.x * 16;
    const int col0 = (blockIdx.y * 4 + wave) * 16;
    const float* W; const float* B; float* O;
    switch (blockIdx.z) {
        case 0:  W = Wq; B = bq; O = Q; break;
        case 1:  W = Wk; B = bk; O = K; break;
        case 2:  W = Wv; B = bv; O = V; break;
        default: W = Ws; B = bs; O = S; break;
    }
    v8f acc = {};
#pragma unroll
    for (int k = 0; k < 16; ++k) {
        const int kk = k * 4 + hi * 2;
        v2f a, b;
        a.x = x[(r0 + n) * DIM + kk];
        a.y = x[(r0 + n) * DIM + kk + 1];
        b.x = W[kk * HDIM + col0 + n];
        b.y = W[(kk + 1) * HDIM + col0 + n];
        acc = __builtin_amdgcn_wmma_f32_16x16x4_f32(false, a, false, b,
                                                    (short)0, acc, false, false);
    }
    const float bias = B[col0 + n];
#pragma unroll
    for (int i = 0; i < 8; ++i)
        O[(size_t)(r0 + i + 8 * hi) * HDIM + col0 + n] = acc[i] + bias;
}

// =====================================================================
// Edge tile loader: 16 edges x 64 feats into padded LDS + the 16 src/dst
// indices. Uses CDNA5 async global->LDS DMA (ASYNCcnt) when available.
// =====================================================================
__device__ __forceinline__ void load_edge_tile(const float* __restrict__ edge_attr,
                                               int e0, float (&sA)[16][68],
                                               const int* __restrict__ esrc,
                                               const int* __restrict__ edst,
                                               int (&sSrc)[16], int (&sDst)[16]) {
#if HAVE_ASYNC_LDS
    {
        // 16 rows x 64 floats = 256 chunks of 16B; 128 threads -> 2 chunks each.
        const int t = threadIdx.x;
#pragma unroll
        for (int c = t; c < 256; c += 128) {
            const int r = c >> 4, j = (c & 15) * 4;
            __builtin_amdgcn_global_load_async_to_lds_b128(
                (__attribute__((address_space(1))) v4i_ld*)(edge_attr + (size_t)(e0 + r) * DIM + j),
                (__attribute__((address_space(3))) v4i_ld*)&sA[r][j],
                0, 0);
        }
    }
#else
    for (int idx = threadIdx.x; idx < 16 * 64; idx += blockDim.x) {
        const int r = idx >> 6, c = idx & 63;
        sA[r][c] = edge_attr[(size_t)(e0 + r) * DIM + c];
    }
#endif
    if (threadIdx.x < 16) sSrc[threadIdx.x] = esrc[e0 + threadIdx.x];
    else if (threadIdx.x < 32) sDst[threadIdx.x - 16] = edst[e0 + threadIdx.x - 16];
#if HAVE_ASYNC_LDS
#if __has_builtin(__builtin_amdgcn_s_wait_asynccnt)
    __builtin_amdgcn_s_wait_asynccnt(0);
#else
    asm volatile("s_wait_asynccnt 0x0" ::: "memory");
#endif
#endif
    __syncthreads();
}

// =====================================================================
// Edge-projection tile (recomputed, never materialized):
// ep = edge_attr_tile[16x64] @ We[:, head*64 .. head*64+63]  -> 16x64 in 4 C-tiles
// =====================================================================
__device__ __forceinline__ void edge_eproj(const float sA[16][68],
                                           const float* __restrict__ We,
                                           int head, int n, int hi, v8f ep[4]) {
    v2f a[16];
#pragma unroll
    for (int k = 0; k < 16; ++k) {
        const int kk = k * 4 + hi * 2;
        a[k].x = sA[n][kk];
        a[k].y = sA[n][kk + 1];
    }
#pragma unroll
    for (int ct = 0; ct < 4; ++ct) {
        const int col0 = head * 64 + ct * 16;
        v8f acc = {};
#pragma unroll
        for (int k = 0; k < 16; ++k) {
            const int kk = k * 4 + hi * 2;
            v2f b;
            b.x = We[kk * HDIM + col0 + n];
            b.y = We[(kk + 1) * HDIM + col0 + n];
            acc = __builtin_amdgcn_wmma_f32_16x16x4_f32(false, a[k], false, b,
                                                        (short)0, acc, false, false);
        }
        ep[ct] = acc;
    }
}

// =====================================================================
// Edge pass 1: logits + per-(dst,head) running max.
// block=128 (wave per head), grid = E/16
// =====================================================================
__global__ void edge_pass1(const float* __restrict__ edge_attr,
                           const int* __restrict__ esrc, const int* __restrict__ edst,
                           const float* __restrict__ We,
                           const float* __restrict__ Q, const float* __restrict__ K,
                           float* __restrict__ logits, unsigned* __restrict__ mkey) {
    __shared__ float sA[16][68];
    __shared__ int sSrc[16], sDst[16];
    const int e0 = blockIdx.x * 16;
    load_edge_tile(edge_attr, e0, sA, esrc, edst, sSrc, sDst);

    const int lane = threadIdx.x & 31;
    const int head = threadIdx.x >> 5;
    const int n = lane & 15, hi = lane >> 4;
    v8f ep[4];
    edge_eproj(sA, We, head, n, hi, ep);

#pragma unroll
    for (int i = 0; i < 8; ++i) {
        const int m = i + 8 * hi;
        const int sr = sSrc[m], dr = sDst[m];
        float part = 0.f;
#pragma unroll
        for (int ct = 0; ct < 4; ++ct) {
            const int col = head * 64 + ct * 16 + n;
            part += Q[(size_t)dr * HDIM + col] * (K[(size_t)sr * HDIM + col] + ep[ct][i]);
        }
#pragma unroll
        for (int off = 1; off < 16; off <<= 1) part += __shfl_xor(part, off, 32);
        if (n == 0) {
            const float lg = part * 0.125f;   // 1/sqrt(64)
            logits[(size_t)(e0 + m) * NH + head] = lg;
            atomicMax(&mkey[(size_t)dr * NH + head], fmax_key(lg));
        }
    }
}

// =====================================================================
// Edge pass 2: p=exp(logit-max); z += p; agg += p*(v+e)  (unnormalized)
// =====================================================================
__global__ void edge_pass2(const float* __restrict__ edge_attr,
                           const int* __restrict__ esrc, const int* __restrict__ edst,
                           const float* __restrict__ We,
                           const float* __restrict__ V,
                           const float* __restrict__ logits,
                           const unsigned* __restrict__ mkey,
                           float* __restrict__ z, float* __restrict__ agg) {
    __shared__ float sA[16][68];
    __shared__ int sSrc[16], sDst[16];
    const int e0 = blockIdx.x * 16;
    load_edge_tile(edge_attr, e0, sA, esrc, edst, sSrc, sDst);

    const int lane = threadIdx.x & 31;
    const int head = threadIdx.x >> 5;
    const int n = lane & 15, hi = lane >> 4;
    v8f ep[4];
    edge_eproj(sA, We, head, n, hi, ep);

#pragma unroll
    for (int i = 0; i < 8; ++i) {
        const int m = i + 8 * hi;
        const int sr = sSrc[m], dr = sDst[m];
        const float lg = logits[(size_t)(e0 + m) * NH + head];
        const float mx = fmax_unkey(mkey[(size_t)dr * NH + head]);
        const float p = __expf(lg - mx);
        if (n == 0) unsafeAtomicAdd(&z[(size_t)dr * NH + head], p);
#pragma unroll
        for (int ct = 0; ct < 4; ++ct) {
            const int col = head * 64 + ct * 16 + n;
            unsafeAtomicAdd(&agg[(size_t)dr * HDIM + col],
                            p * (V[(size_t)sr * HDIM + col] + ep[ct][i]));
        }
    }
}

// =====================================================================
// h = agg/z + skip ; accumulate per-channel BN sums
// =====================================================================
__global__ void combine_bnstats(const float* __restrict__ agg, const float* __restrict__ z,
                                const float* __restrict__ S, float* __restrict__ H,
                                float* __restrict__ bnsum, float* __restrict__ bnsq,
                                int rowsPer) {
    const int c = threadIdx.x;
    const int r0 = blockIdx.x * rowsPer;
    float s = 0.f, sq = 0.f;
    for (int r = r0; r < r0 + rowsPer; ++r) {
        const float zz = z[(size_t)r * NH + (c >> 6)];
        const float h = S[(size_t)r * HDIM + c] +
                        (zz > 0.f ? agg[(size_t)r * HDIM + c] / zz : 0.f);
        H[(size_t)r * HDIM + c] = h;
        s += h; sq += h * h;
    }
    unsafeAtomicAdd(&bnsum[c], s);
    unsafeAtomicAdd(&bnsq[c], sq);
}

__global__ void bn_finalize(const float* __restrict__ bnsum, const float* __restrict__ bnsq,
                            const float* __restrict__ gamma, const float* __restrict__ beta,
                            float* __restrict__ affA, float* __restrict__ affD) {
    const int c = threadIdx.x;
    const float mu = bnsum[c] * (1.f / N_NODES);
    const float var = bnsq[c] * (1.f / N_NODES) - mu * mu;
    const float rs = rsqrtf(var + BN_EPS);
    const float aa = gamma[c] * rs;
    affA[c] = aa;
    affD[c] = beta[c] - mu * aa;
}

// =====================================================================
// x_next = relu( BN(h) @ Wt + bt )   [N,256]@[256,64]; BN folded into A load
// =====================================================================
__global__ void xnext_gemm(const float* __restrict__ H,
                           const float* __restrict__ affA, const float* __restrict__ affD,
                           const float* __restrict__ Wt, const float* __restrict__ bt,
                           float* __restrict__ xout) {
    const int lane = threadIdx.x & 31;
    const int wave = threadIdx.x >> 5;
    const int n = lane & 15, hi = lane >> 4;
    const int r0 = blockIdx.x * 16;
    const int col0 = wave * 16;
    v8f acc = {};
    for (int k = 0; k < 64; ++k) {
        const int kk = k * 4 + hi * 2;
        v2f a, b;
        a.x = H[(size_t)(r0 + n) * HDIM + kk] * affA[kk] + affD[kk];
        a.y = H[(size_t)(r0 + n) * HDIM + kk + 1] * affA[kk + 1] + affD[kk + 1];
        b.x = Wt[kk * DIM + col0 + n];
        b.y = Wt[(kk + 1) * DIM + col0 + n];
        acc = __builtin_amdgcn_wmma_f32_16x16x4_f32(false, a, false, b,
                                                    (short)0, acc, false, false);
    }
    const float bias = bt[col0 + n];
#pragma unroll
    for (int i = 0; i < 8; ++i) {
        const float v = acc[i] + bias;
        xout[(size_t)(r0 + i + 8 * hi) * DIM + col0 + n] = v > 0.f ? v : 0.f;
    }
}

// =====================================================================
// Pooling: per-graph max (uint atomicMax valid since x>=0 post-ReLU) + sum
// =====================================================================
__global__ void pool_kernel(const float* __restrict__ xv, const int* __restrict__ batch,
                            unsigned* __restrict__ gmx, float* __restrict__ gav) {
    const int idx = blockIdx.x * blockDim.x + threadIdx.x;   // over N*64
    const int node = idx >> 6, c = idx & 63;
    const int g = batch[node];
    const float v = xv[idx];
    atomicMax(&gmx[g * 64 + c], __float_as_uint(v));
    unsafeAtomicAdd(&gav[g * 64 + c], v);
}

__global__ void rep_acc(const unsigned* __restrict__ gmx, const float* __restrict__ gav,
                        const int* __restrict__ counts, float* __restrict__ rep) {
    const int g = blockIdx.x, c = threadIdx.x;  // 128 threads
    float add;
    if (c < 64) add = __uint_as_float(gmx[g * 64 + c]);
    else        add = gav[g * 64 + (c - 64)] / (float)counts[g];
    rep[g * 128 + c] += add;
}

// =====================================================================
// MLP head (tiny; plain VALU)
// =====================================================================
__global__ void head1(const float* __restrict__ rep, const float* __restrict__ W1,
                      const float* __restrict__ b1, float* __restrict__ t1) {
    const int g = blockIdx.x, c = threadIdx.x;  // 256
    float s = b1[c];
    for (int k = 0; k < 2 * DIM; ++k) s += rep[g * 128 + k] * W1[k * DENSE + c];
    t1[g * DENSE + c] = s > 0.f ? s : 0.f;
}
__global__ void head2(const float* __restrict__ t1, const float* __restrict__ W2,
                      const float* __restrict__ b2, float* __restrict__ t2) {
    const int g = blockIdx.x, c = threadIdx.x;  // 128
    float s = b2[c];
    for (int k = 0; k < DENSE; ++k) s += t1[g * DENSE + k] * W2[k * (DENSE / 2) + c];
    t2[g * (DENSE / 2) + c] = s > 0.f ? s : 0.f;
}
__global__ void head3(const float* __restrict__ t2, const float* __restrict__ W3,
                      const float* __restrict__ b3, float* __restrict__ out) {
    const int g = threadIdx.x;  // 64
    float s = b3[0];
    for (int k = 0; k < DENSE / 2; ++k) s += t2[g * (DENSE / 2) + k] * W3[k];
    out[g] = s;
}

// =====================================================================
extern "C" void kernel_launch(void* const* d_in, const int* in_sizes, int n_in,
                              void* d_out, int out_size, void* d_ws, size_t ws_size,
                              hipStream_t stream) {
    (void)in_sizes; (void)n_in; (void)out_size; (void)ws_size;
    const float* x_in      = (const float*)d_in[0];
    const float* edge_attr = (const float*)d_in[1];
    const int*   eidx      = (const int*)d_in[2];
    const int*   batch     = (const int*)d_in[3];
    const float* Wq = (const float*)d_in[4];  const float* bq = (const float*)d_in[5];
    const float* Wk = (const float*)d_in[6];  const float* bk = (const float*)d_in[7];
    const float* Wv = (const float*)d_in[8];  const float* bv = (const float*)d_in[9];
    const float* We = (const float*)d_in[10];
    const float* Wsk = (const float*)d_in[11]; const float* bsk = (const float*)d_in[12];
    const float* bn_g = (const float*)d_in[13]; const float* bn_b = (const float*)d_in[14];
    const float* Wt = (const float*)d_in[15]; const float* bt = (const float*)d_in[16];
    const float* W1 = (const float*)d_in[17]; const float* b1 = (const float*)d_in[18];
    const float* W2 = (const float*)d_in[19]; const float* b2 = (const float*)d_in[20];
    const float* W3 = (const float*)d_in[21]; const float* b3 = (const float*)d_in[22];
    const int* esrc = eidx;
    const int* edst = eidx + N_EDGES;

    float* ws = (float*)d_ws;
    size_t o = 0;
    float* Qb  = ws + o; o += (size_t)N_NODES * HDIM;   // also reused as H buffer
    float* Kb  = ws + o; o += (size_t)N_NODES * HDIM;
    float* Vb  = ws + o; o += (size_t)N_NODES * HDIM;
    float* Sb  = ws + o; o += (size_t)N_NODES * HDIM;
    float* agg = ws + o; o += (size_t)N_NODES * HDIM;
    float* xbuf = ws + o; o += (size_t)N_NODES * DIM;
    float* logits = ws + o; o += (size_t)N_EDGES * NH;
    unsigned* mkey = (unsigned*)(ws + o); o += (size_t)N_NODES * NH;
    float* z = ws + o; o += (size_t)N_NODES * NH;
    float* bnsum = ws + o; o += HDIM;
    float* bnsq  = ws + o; o += HDIM;
    float* affA  = ws + o; o += HDIM;
    float* affD  = ws + o; o += HDIM;
    unsigned* gmx = (unsigned*)(ws + o); o += NG * 64;
    float* gav = ws + o; o += NG * 64;
    int* counts = (int*)(ws + o); o += NG;
    float* rep = ws + o; o += NG * 128;
    float* t1 = ws + o; o += NG * DENSE;
    float* t2 = ws + o; o += NG * (DENSE / 2);

    auto fill = [&](void* p, unsigned v, size_t n) {
        fill_u32<<<dim3((unsigned)((n + 255) / 256)), dim3(256), 0, stream>>>((unsigned*)p, v, (int)n);
    };

    fill(counts, 0u, NG);
    fill(rep, 0u, (size_t)NG * 128);
    count_kernel<<<dim3((N_NODES + 255) / 256), dim3(256), 0, stream>>>(batch, counts);

    for (int i = 0; i < NL; ++i) {
        const float* xcur = (i == 0) ? x_in : xbuf;
        fill(agg, 0u, (size_t)N_NODES * HDIM);
        fill(z, 0u, (size_t)N_NODES * NH);
        fill(mkey, NEG_INF_KEY, (size_t)N_NODES * NH);
        fill(bnsum, 0u, 2 * HDIM);          // bnsum + bnsq are contiguous
        fill(gmx, 0u, NG * 64);
        fill(gav, 0u, NG * 64);

        qkvs_gemm<<<dim3(N_NODES / 16, 4, 4), dim3(128), 0, stream>>>(
            xcur,
            Wq + (size_t)i * DIM * HDIM, bq + (size_t)i * HDIM,
            Wk + (size_t)i * DIM * HDIM, bk + (size_t)i * HDIM,
            Wv + (size_t)i * DIM * HDIM, bv + (size_t)i * HDIM,
            Wsk + (size_t)i * DIM * HDIM, bsk + (size_t)i * HDIM,
            Qb, Kb, Vb, Sb);

        const float* Wei = We + (size_t)i * DIM * HDIM;
        edge_pass1<<<dim3(N_EDGES / 16), dim3(128), 0, stream>>>(
            edge_attr, esrc, edst, Wei, Qb, Kb, logits, mkey);
        edge_pass2<<<dim3(N_EDGES / 16), dim3(128), 0, stream>>>(
            edge_attr, esrc, edst, Wei, Vb, logits, mkey, z, agg);

        combine_bnstats<<<dim3(200), dim3(256), 0, stream>>>(
            agg, z, Sb, Qb /* H reuses Q */, bnsum, bnsq, N_NODES / 200);
        bn_finalize<<<dim3(1), dim3(256), 0, stream>>>(
            bnsum, bnsq, bn_g + (size_t)i * HDIM, bn_b + (size_t)i * HDIM, affA, affD);

        xnext_gemm<<<dim3(N_NODES / 16), dim3(128), 0, stream>>>(
            Qb, affA, affD, Wt + (size_t)i * HDIM * DIM, bt + (size_t)i * DIM, xbuf);

        pool_kernel<<<dim3(N_NODES * DIM / 256), dim3(256), 0, stream>>>(xbuf, batch, gmx, gav);
        rep_acc<<<dim3(NG), dim3(128), 0, stream>>>(gmx, gav, counts, rep);
    }

    head1<<<dim3(NG), dim3(DENSE), 0, stream>>>(rep, W1, b1, t1);
    head2<<<dim3(NG), dim3(DENSE / 2), 0, stream>>>(t1, W2, b2, t2);
    head3<<<dim3(1), dim3(NG), 0, stream>>>(t2, W3, b3, (float*)d_out);
}